// Memory_67250597920918
// MI455X (gfx1250) — compile-verified
//
#include <hip/hip_runtime.h>

// RWKV WKV scan, B=8, T=2048, H=2048, fp32 (matches reference setup_inputs).
// Memory-bound streaming recurrence -> chunk-parallel 3-phase scan.
// No matrix op exists in this workload, so no WMMA; CDNA5 leverage is the
// memory path: NT temporal hints, global_prefetch_b8, B128 vector accesses.

typedef float v4f __attribute__((ext_vector_type(4)));

constexpr unsigned Bc     = 8;
constexpr unsigned Tc     = 2048;
constexpr unsigned Hc     = 2048;
constexpr unsigned CHUNKS = 16;
constexpr unsigned Lc     = Tc / CHUNKS;   // 128 timesteps per chunk
constexpr unsigned VEC    = 4;             // channels per thread (float4)
constexpr unsigned HV     = Hc / VEC;      // 512 vector-channels per (b)
constexpr unsigned HV_SH  = 9;             // log2(HV)
constexpr unsigned B_SH   = 3;             // log2(Bc)
constexpr unsigned PF     = 32;            // prefetch distance in timesteps

static_assert(Tc % CHUNKS == 0, "chunking must divide T");
static_assert(Hc % VEC == 0, "vector width must divide H");
static_assert((1u << HV_SH) == HV && (1u << B_SH) == Bc, "shift constants");
static_assert(PF < Lc && (Lc - PF) % 4 == 0 && PF % 4 == 0, "clean loop split");

// One timestep of the aggregate-only recurrence (phase 1).
template <bool PREFETCH>
__device__ __forceinline__ void step_agg(const float*& kq, const float*& vq,
                                         const v4f& decay, v4f& a, v4f& s)
{
    if (PREFETCH) {
        __builtin_prefetch(kq + (size_t)PF * Hc, 0, 0);  // global_prefetch_b8
        __builtin_prefetch(vq + (size_t)PF * Hc, 0, 0);
    }
    v4f kk = __builtin_nontemporal_load((const v4f*)kq);
    v4f vv = __builtin_nontemporal_load((const v4f*)vq);
    #pragma unroll
    for (unsigned i = 0; i < VEC; ++i) {
        float imp = __expf(kk[i]);                       // hot path: v_exp_f32
        a[i] = fmaf(decay[i], a[i], imp * vv[i]);
        s[i] = fmaf(decay[i], s[i], imp);
    }
    kq += Hc; vq += Hc;
}

// One timestep of the full recurrence with output (phase 3 / fallback).
template <bool PREFETCH>
__device__ __forceinline__ void step_out(const float*& kq, const float*& vq,
                                         float*& oq, const v4f& decay,
                                         const v4f& bonus, v4f& a, v4f& s)
{
    if (PREFETCH) {
        __builtin_prefetch(kq + (size_t)PF * Hc, 0, 0);
        __builtin_prefetch(vq + (size_t)PF * Hc, 0, 0);
    }
    v4f kk = __builtin_nontemporal_load((const v4f*)kq);
    v4f vv = __builtin_nontemporal_load((const v4f*)vq);
    v4f wkv;
    #pragma unroll
    for (unsigned i = 0; i < VEC; ++i) {
        float imp = __expf(kk[i]);
        float w   = bonus[i] * imp;
        wkv[i]    = __fdividef(fmaf(w, vv[i], a[i]), s[i] + w); // off the chain
        a[i] = fmaf(decay[i], a[i], imp * vv[i]);               // chain: 1 FMA
        s[i] = fmaf(decay[i], s[i], imp);                       // chain: 1 FMA
    }
    __builtin_nontemporal_store(wkv, (v4f*)oq);
    kq += Hc; vq += Hc; oq += Hc;
}

// ---------------------------------------------------------------------------
// Phase 1: per (chunk, b, h4) compute chunk-local aggregates from zero state.
// Only chunks 0..CHUNKS-2 (last chunk's aggregate is never consumed).
// ---------------------------------------------------------------------------
__global__ __launch_bounds__(256) void wkv_phase1(
    const float* __restrict__ key, const float* __restrict__ value,
    const float* __restrict__ time_decay,
    float* __restrict__ aggA, float* __restrict__ aggB)
{
    const unsigned gid = blockIdx.x * blockDim.x + threadIdx.x;  // (c, b, hv)
    const unsigned hv  = gid & (HV - 1);
    const unsigned cb  = gid >> HV_SH;
    const unsigned b   = cb & (Bc - 1);
    const unsigned c   = cb >> B_SH;
    const unsigned h0  = hv * VEC;

    v4f decay;
    #pragma unroll
    for (unsigned i = 0; i < VEC; ++i)
        decay[i] = expf(-expf(time_decay[h0 + i]));  // precise, once per thread

    const size_t base = ((size_t)b * Tc + (size_t)c * Lc) * Hc + h0;
    const float* kq = key + base;
    const float* vq = value + base;

    v4f a = 0.0f, s = 0.0f;
    #pragma unroll 4
    for (unsigned t = 0; t < Lc - PF; ++t)
        step_agg<true>(kq, vq, decay, a, s);
    #pragma unroll 4
    for (unsigned t = 0; t < PF; ++t)
        step_agg<false>(kq, vq, decay, a, s);

    const size_t o = (size_t)cb * Hc + h0;           // [c][b][h] flat
    *(v4f*)(aggA + o) = a;
    *(v4f*)(aggB + o) = s;
}

// ---------------------------------------------------------------------------
// Phase 2: per channel (b,h), exclusive scan over chunk aggregates, in place.
// incoming[c] = dL * incoming[c-1] + agg[c-1],  dL = decay^Lc.
// ---------------------------------------------------------------------------
__global__ __launch_bounds__(256) void wkv_phase2(
    const float* __restrict__ time_decay,
    float* __restrict__ aggA, float* __restrict__ aggB)
{
    const unsigned ch = blockIdx.x * blockDim.x + threadIdx.x;  // b*Hc + h
    const unsigned h  = ch & (Hc - 1);
    const float dL = expf(-expf(time_decay[h]) * (float)Lc);    // decay^Lc

    float curA = 0.0f, curB = 0.0f;
    #pragma unroll
    for (unsigned c = 0; c < CHUNKS; ++c) {
        const size_t idx = (size_t)c * (Bc * Hc) + ch;
        float tA = 0.0f, tB = 0.0f;
        if (c < CHUNKS - 1) { tA = aggA[idx]; tB = aggB[idx]; } // slot 15 unwritten
        aggA[idx] = curA;
        aggB[idx] = curB;
        curA = fmaf(dL, curA, tA);
        curB = fmaf(dL, curB, tB);
    }
}

// ---------------------------------------------------------------------------
// Phase 3: per (chunk, b, h4), replay the chunk with the correct incoming
// state and emit wkv_t = (a + bonus*imp*v) / (b + bonus*imp).
// ---------------------------------------------------------------------------
__global__ __launch_bounds__(256) void wkv_phase3(
    const float* __restrict__ key, const float* __restrict__ value,
    const float* __restrict__ time_decay, const float* __restrict__ time_first,
    const float* __restrict__ inA, const float* __restrict__ inB,
    float* __restrict__ out)
{
    const unsigned gid = blockIdx.x * blockDim.x + threadIdx.x;  // (c, b, hv)
    const unsigned hv  = gid & (HV - 1);
    const unsigned cb  = gid >> HV_SH;
    const unsigned b   = cb & (Bc - 1);
    const unsigned c   = cb >> B_SH;
    const unsigned h0  = hv * VEC;

    v4f decay, bonus;
    #pragma unroll
    for (unsigned i = 0; i < VEC; ++i) {
        decay[i] = expf(-expf(time_decay[h0 + i]));
        bonus[i] = expf(time_first[h0 + i]);
    }

    const size_t o = (size_t)cb * Hc + h0;
    v4f a = *(const v4f*)(inA + o);
    v4f s = *(const v4f*)(inB + o);

    const size_t base = ((size_t)b * Tc + (size_t)c * Lc) * Hc + h0;
    const float* kq = key + base;
    const float* vq = value + base;
    float*       oq = out + base;

    #pragma unroll 4
    for (unsigned t = 0; t < Lc - PF; ++t)
        step_out<true>(kq, vq, oq, decay, bonus, a, s);
    #pragma unroll 4
    for (unsigned t = 0; t < PF; ++t)
        step_out<false>(kq, vq, oq, decay, bonus, a, s);
}

// ---------------------------------------------------------------------------
// Fallback: single-pass scan (used only if ws_size < 2 MB; deterministic).
// ---------------------------------------------------------------------------
__global__ __launch_bounds__(256) void wkv_single(
    const float* __restrict__ key, const float* __restrict__ value,
    const float* __restrict__ time_decay, const float* __restrict__ time_first,
    float* __restrict__ out)
{
    const unsigned gid = blockIdx.x * blockDim.x + threadIdx.x;  // (b, hv)
    const unsigned hv  = gid & (HV - 1);
    const unsigned b   = gid >> HV_SH;
    const unsigned h0  = hv * VEC;

    v4f decay, bonus;
    #pragma unroll
    for (unsigned i = 0; i < VEC; ++i) {
        decay[i] = expf(-expf(time_decay[h0 + i]));
        bonus[i] = expf(time_first[h0 + i]);
    }

    const size_t base = (size_t)b * Tc * Hc + h0;
    const float* kq = key + base;
    const float* vq = value + base;
    float*       oq = out + base;

    v4f a = 0.0f, s = 0.0f;
    #pragma unroll 4
    for (unsigned t = 0; t < Tc - PF; ++t)
        step_out<true>(kq, vq, oq, decay, bonus, a, s);
    #pragma unroll 4
    for (unsigned t = 0; t < PF; ++t)
        step_out<false>(kq, vq, oq, decay, bonus, a, s);
}

extern "C" void kernel_launch(void* const* d_in, const int* in_sizes, int n_in,
                              void* d_out, int out_size, void* d_ws, size_t ws_size,
                              hipStream_t stream) {
    (void)in_sizes; (void)n_in; (void)out_size;
    const float* key   = (const float*)d_in[0];
    const float* value = (const float*)d_in[1];
    const float* td    = (const float*)d_in[2];
    const float* tf    = (const float*)d_in[3];
    float*       out   = (float*)d_out;

    const size_t aggElems = (size_t)CHUNKS * Bc * Hc;          // 262144
    const size_t need     = 2 * aggElems * sizeof(float);      // 2 MB

    if (ws_size >= need) {
        float* aggA = (float*)d_ws;
        float* aggB = aggA + aggElems;
        // Phase 1: (CHUNKS-1)*B*HV threads = 61440 -> 240 blocks
        wkv_phase1<<<dim3((CHUNKS - 1) * Bc * HV / 256), dim3(256), 0, stream>>>(
            key, value, td, aggA, aggB);
        // Phase 2: B*H threads = 16384 -> 64 blocks
        wkv_phase2<<<dim3(Bc * Hc / 256), dim3(256), 0, stream>>>(td, aggA, aggB);
        // Phase 3: CHUNKS*B*HV threads = 65536 -> 256 blocks (2048 wave32s)
        wkv_phase3<<<dim3(CHUNKS * Bc * HV / 256), dim3(256), 0, stream>>>(
            key, value, td, tf, aggA, aggB, out);
    } else {
        wkv_single<<<dim3(Bc * HV / 256), dim3(256), 0, stream>>>(key, value, td, tf, out);
    }
}